// Word2VecKMerEmb_14559939134042
// MI455X (gfx1250) — compile-verified
//
#include <hip/hip_runtime.h>
#include <hip/hip_bf16.h>
#include <math.h>

#define KN 4096
#define CN 10
#define DN 128
#define BN 1000
#define BPAD 1008   // 63 * 16
#define LN 512
#define TPAD 132    // embs LDS tile row pitch (floats): 132 mod 64 = 4 -> conflict-free
#define PPAD 68     // pc LDS tile row pitch (floats): 68 mod 64 = 4 -> conflict-free

typedef __attribute__((ext_vector_type(2))) float v2f;
typedef __attribute__((ext_vector_type(8))) float v8f;

// ---------------- row norms: sqn[k] = sum_d embs[k][d]^2 ----------------
__global__ void k_row_norms(const float* __restrict__ embs, float* __restrict__ sqn) {
    int row = blockIdx.x * blockDim.x + threadIdx.x;
    if (row >= KN) return;
    const float4* e = reinterpret_cast<const float4*>(embs + (size_t)row * DN);
    float s = 0.f;
#pragma unroll
    for (int i = 0; i < DN / 4; ++i) {
        float4 v = e[i];
        s += v.x * v.x + v.y * v.y + v.z * v.z + v.w * v.w;
    }
    sqn[row] = s;
}

// ---------------- zero-padded softmax weights: Wp[16][128] ----------------
__global__ void k_pad_w(const float* __restrict__ W, float* __restrict__ Wp) {
    int idx = blockIdx.x * blockDim.x + threadIdx.x;   // 0..2047
    int n = idx >> 7, k = idx & 127;
    Wp[idx] = (n < CN) ? W[n * DN + k] : 0.f;
}

// ---------------- read_emb[b][d] = sum_l embs[reads[b][l]][d]  (rows >= BN zeroed) --------
__global__ void k_read_emb(const int* __restrict__ reads, const float* __restrict__ embs,
                           float* __restrict__ read_emb) {
    __shared__ int kml[LN];
    int b = blockIdx.x;
    int tid = threadIdx.x;               // 128 threads, one per dim
    if (b >= BN) { read_emb[(size_t)b * DN + tid] = 0.f; return; }
    for (int i = tid; i < LN; i += 128) kml[i] = reads[b * LN + i];
    __syncthreads();
    float acc = 0.f;
    for (int l = 0; l < LN; ++l) acc += embs[(size_t)kml[l] * DN + tid];
    read_emb[(size_t)b * DN + tid] = acc;
}

// ---------------- supervised: logits via fp32 WMMA, per-row NLL ----------------
__global__ void k_sup_loss(const float* __restrict__ read_emb,   // [BPAD][DN]
                           const float* __restrict__ Wp,         // [16][DN], zero padded
                           const int* __restrict__ labels,
                           float* __restrict__ loss_sup) {
    __shared__ float lt[16 * 17];
    int lane   = threadIdx.x & 31;
    int laneLo = lane & 15;
    int hi     = lane >> 4;              // 0: K pair {0,1}, 1: K pair {2,3}
    int rowBase = blockIdx.x * 16;
    const float* arow = read_emb + (size_t)(rowBase + laneLo) * DN;
    const float* brow = Wp + (size_t)laneLo * DN;
    v8f acc = {};
#pragma unroll 8
    for (int it = 0; it < DN / 4; ++it) {
        int kb = it * 4 + 2 * hi;
        v2f a = *reinterpret_cast<const v2f*>(arow + kb);
        v2f b = *reinterpret_cast<const v2f*>(brow + kb);
        acc = __builtin_amdgcn_wmma_f32_16x16x4_f32(false, a, false, b,
                                                    (short)0, acc, false, false);
    }
#pragma unroll
    for (int r = 0; r < 8; ++r)
        lt[(r + 8 * hi) * 17 + laneLo] = acc[r];
    __syncthreads();
    if (lane < 16) {
        int row = rowBase + lane;
        if (row < BN) {
            float m = -1e30f;
            for (int c = 0; c < CN; ++c) m = fmaxf(m, lt[lane * 17 + c]);
            float s = 0.f;
            for (int c = 0; c < CN; ++c) s += expf(lt[lane * 17 + c] - m);
            float lse = m + logf(s);
            int lab = labels[row];
            loss_sup[row] = -(lt[lane * 17 + lab] - lse);
        }
    }
}

// ---------------- unsupervised: 64x64 Gram tile per block, fused epilogue ----------------
// All global traffic (embs tiles + pair_counts tile) staged via async DMA to LDS.
__global__ void __launch_bounds__(256)
k_unsup(const float* __restrict__ embs, const float* __restrict__ sqn,
        const float* __restrict__ pc, float* __restrict__ partials) {
    __shared__ float As[64 * TPAD];
    __shared__ float Bs[64 * TPAD];
    __shared__ float Ps[64 * PPAD];
    __shared__ float red[256];
    int tid = threadIdx.x;
    int rowBase = blockIdx.y * 64;
    int colBase = blockIdx.x * 64;

    // batch 1: embedding tiles (16 async b128 per thread)
#pragma unroll
    for (int i = 0; i < 8; ++i) {
        int idx = tid + i * 256;         // 0..2047 float4 slots
        int r = idx >> 5;                // 32 float4 per row
        int q = idx & 31;
        const float* ga = embs + (size_t)(rowBase + r) * DN + q * 4;
        const float* gb = embs + (size_t)(colBase + r) * DN + q * 4;
        unsigned la = (unsigned)(uintptr_t)(&As[r * TPAD + q * 4]);
        unsigned lb = (unsigned)(uintptr_t)(&Bs[r * TPAD + q * 4]);
        asm volatile("global_load_async_to_lds_b128 %0, %1, off"
                     :: "v"(la), "v"(ga) : "memory");
        asm volatile("global_load_async_to_lds_b128 %0, %1, off"
                     :: "v"(lb), "v"(gb) : "memory");
    }
    // batch 2: pair_counts tile (4 async b128 per thread), overlaps the WMMA loop
#pragma unroll
    for (int i = 0; i < 4; ++i) {
        int idx = tid + i * 256;         // 0..1023 float4 slots, 16 per row
        int r = idx >> 4;
        int q = idx & 15;
        const float* gp = pc + (size_t)(rowBase + r) * KN + colBase + q * 4;
        unsigned lp = (unsigned)(uintptr_t)(&Ps[r * PPAD + q * 4]);
        asm volatile("global_load_async_to_lds_b128 %0, %1, off"
                     :: "v"(lp), "v"(gp) : "memory");
    }
    // wait for batch 1 only (<=4 outstanding => the 16 embs asyncs completed, in order)
    asm volatile("s_wait_asynccnt 0x4" ::: "memory");
    __syncthreads();

    int wave = tid >> 5;
    int lane = tid & 31;
    int laneLo = lane & 15;
    int hi = lane >> 4;
    // 16 tiles (4x4); wave w owns tm = w>>1, tn in {2*(w&1), 2*(w&1)+1}; A shared
    int tm  = wave >> 1;
    int tn0 = (wave & 1) * 2;
    const float* Arow = &As[(tm * 16 + laneLo) * TPAD];
    const float* B0   = &Bs[(tn0 * 16 + laneLo) * TPAD];
    const float* B1   = &Bs[((tn0 + 1) * 16 + laneLo) * TPAD];
    v8f acc0 = {}, acc1 = {};
#pragma unroll 8
    for (int it = 0; it < DN / 4; ++it) {
        int kb = it * 4 + 2 * hi;        // A 16x4 fp32: lanes0-15 K{0,1}, lanes16-31 K{2,3}
        v2f a  = *reinterpret_cast<const v2f*>(Arow + kb);
        v2f b0 = *reinterpret_cast<const v2f*>(B0 + kb);
        v2f b1 = *reinterpret_cast<const v2f*>(B1 + kb);
        acc0 = __builtin_amdgcn_wmma_f32_16x16x4_f32(false, a, false, b0,
                                                     (short)0, acc0, false, false);
        acc1 = __builtin_amdgcn_wmma_f32_16x16x4_f32(false, a, false, b1,
                                                     (short)0, acc1, false, false);
    }
    // pc tile DMA must be complete and visible to all waves
    asm volatile("s_wait_asynccnt 0x0" ::: "memory");
    __syncthreads();

    float lsum = 0.f;
#pragma unroll
    for (int s = 0; s < 2; ++s) {
        const v8f& acc = s ? acc1 : acc0;
        int lj = (tn0 + s) * 16 + laneLo;
        float sj = sqn[colBase + lj];
#pragma unroll
        for (int r = 0; r < 8; ++r) {
            int li = tm * 16 + r + 8 * hi;
            float p  = Ps[li * PPAD + lj];
            float sq = sqn[rowBase + li] + sj - 2.f * acc[r];
            sq = fmaxf(sq, 0.f);
            if (p != 0.f) {
                float d = sqrtf(sq);
                lsum += p * d + expf(-d);   // = -(term); scale applied at finalize
            }
        }
    }
    red[tid] = lsum;
    __syncthreads();
    for (int s = 128; s > 0; s >>= 1) {
        if (tid < s) red[tid] += red[tid + s];
        __syncthreads();
    }
    if (tid == 0) partials[blockIdx.y * gridDim.x + blockIdx.x] = red[0];
}

// ---------------- final deterministic reduction ----------------
__global__ void k_finalize(const float* __restrict__ loss_sup,
                           const float* __restrict__ partials,
                           const float* __restrict__ delta,
                           float* __restrict__ out) {
    __shared__ float red[256];
    int tid = threadIdx.x;
    float s = 0.f;
    for (int i = tid; i < BN; i += 256) s += loss_sup[i];
    red[tid] = s;
    __syncthreads();
    for (int st = 128; st > 0; st >>= 1) { if (tid < st) red[tid] += red[tid + st]; __syncthreads(); }
    float sup = red[0];
    __syncthreads();
    float u = 0.f;
    for (int i = tid; i < 4096; i += 256) u += partials[i];
    red[tid] = u;
    __syncthreads();
    for (int st = 128; st > 0; st >>= 1) { if (tid < st) red[tid] += red[tid + st]; __syncthreads(); }
    if (tid == 0) {
        float unsup = red[0] * (1.0f / (4096.0f * 4096.0f));  // sum(pc*d + e^-d) * scale
        float dl = delta[0];
        out[0] = dl * sup + (1.f - dl) * unsup;
    }
}

extern "C" void kernel_launch(void* const* d_in, const int* in_sizes, int n_in,
                              void* d_out, int out_size, void* d_ws, size_t ws_size,
                              hipStream_t stream) {
    const float* pair_counts = (const float*)d_in[0];
    const int*   reads       = (const int*)d_in[1];
    const int*   labels      = (const int*)d_in[2];
    const float* delta       = (const float*)d_in[3];
    const float* embs        = (const float*)d_in[4];
    const float* W           = (const float*)d_in[5];
    (void)in_sizes; (void)n_in; (void)out_size; (void)ws_size;

    char* ws = (char*)d_ws;
    float* sqn      = (float*)(ws);                    // 16 KB  @ 0
    float* read_emb = (float*)(ws + 16384);            // 1008*128*4 = 516096 B
    float* Wp       = (float*)(ws + 16384 + 516096);   // 8 KB
    float* loss_sup = (float*)(ws + 16384 + 516096 + 8192);            // 4 KB (pad 8)
    float* partials = (float*)(ws + 16384 + 516096 + 8192 + 8192);     // 16 KB

    k_row_norms<<<KN / 256, 256, 0, stream>>>(embs, sqn);
    k_pad_w<<<8, 256, 0, stream>>>(W, Wp);
    k_read_emb<<<BPAD, 128, 0, stream>>>(reads, embs, read_emb);
    k_sup_loss<<<BPAD / 16, 32, 0, stream>>>(read_emb, Wp, labels, loss_sup);
    k_unsup<<<dim3(64, 64), 256, 0, stream>>>(embs, sqn, pair_counts, partials);
    k_finalize<<<1, 256, 0, stream>>>(loss_sup, partials, delta, (float*)d_out);
}